// VS_LightGCN_28501402976881
// MI455X (gfx1250) — compile-verified
//
#include <hip/hip_runtime.h>

typedef __attribute__((ext_vector_type(2))) float v2f;
typedef __attribute__((ext_vector_type(8))) float v8f;

#define DIMF 64

// cur = concat(user_emb, item_emb); acc = cur   (e0 term of the mean)
__global__ void lgcn_init_kernel(float* __restrict__ cur, float* __restrict__ acc,
                                 const float* __restrict__ ue, const float* __restrict__ ie,
                                 int NU, long total4) {
  long t = (long)blockIdx.x * blockDim.x + threadIdx.x;
  if (t >= total4) return;
  long o = t * 4;                       // flat float offset, 16B aligned
  int n = (int)(o >> 6);
  int d = (int)(o & 63);
  const float4 v = (n < NU)
      ? *(const float4*)(ue + (size_t)n * DIMF + d)
      : *(const float4*)(ie + (size_t)(n - NU) * DIMF + d);
  *(float4*)(cur + o) = v;
  *(float4*)(acc + o) = v;
}

// nxt = ALPHA * emb0   (scatter-adds land on top of this)
__global__ void lgcn_layerinit_kernel(float* __restrict__ nxt,
                                      const float* __restrict__ ue0, const float* __restrict__ ie0,
                                      int NU, long total4) {
  long t = (long)blockIdx.x * blockDim.x + threadIdx.x;
  if (t >= total4) return;
  long o = t * 4;
  int n = (int)(o >> 6);
  int d = (int)(o & 63);
  const float4 v = (n < NU)
      ? *(const float4*)(ue0 + (size_t)n * DIMF + d)
      : *(const float4*)(ie0 + (size_t)(n - NU) * DIMF + d);
  float4 r; r.x = 0.5f * v.x; r.y = 0.5f * v.y; r.z = 0.5f * v.z; r.w = 0.5f * v.w;
  *(float4*)(nxt + o) = r;
}

// COO SpMM scatter: 16 lanes per edge, float4 gather (L2-resident table),
// NT loads for single-use edge streams, returnless f32 atomics into nxt.
__global__ void lgcn_spmm_kernel(float* __restrict__ nxt, const float* __restrict__ cur,
                                 const int* __restrict__ rows, const int* __restrict__ cols,
                                 const float* __restrict__ vals, int nnz) {
  long t = (long)blockIdx.x * blockDim.x + threadIdx.x;
  int e = (int)(t >> 4);
  if (e >= nnz) return;
  int d0 = ((int)t & 15) * 4;
  int r = __builtin_nontemporal_load(rows + e);
  int c = __builtin_nontemporal_load(cols + e);
  float v = __builtin_nontemporal_load(vals + e);
  const float4 s = *(const float4*)(cur + (size_t)c * DIMF + d0);
  float* dst = nxt + (size_t)r * DIMF + d0;
  atomicAdd(dst + 0, v * s.x);
  atomicAdd(dst + 1, v * s.y);
  atomicAdd(dst + 2, v * s.z);
  atomicAdd(dst + 3, v * s.w);
}

// acc += w * nxt
__global__ void lgcn_accum_kernel(float* __restrict__ acc, const float* __restrict__ nxt,
                                  float w, long total4) {
  long t = (long)blockIdx.x * blockDim.x + threadIdx.x;
  if (t >= total4) return;
  long o = t * 4;
  float4 a = *(float4*)(acc + o);
  const float4 b = *(const float4*)(nxt + o);
  a.x += w * b.x; a.y += w * b.y; a.z += w * b.z; a.w += w * b.w;
  *(float4*)(acc + o) = a;
}

// gamma via WMMA: one wave per 16 pairs. G = U(16x64) x V^T(64x16) using 16x
// V_WMMA_F32_16X16X4_F32 over K, diagonal extracted through LDS.
// A 16x4 f32 layout: lane l holds row (l&15), K = k0 + 2*(l>>4) .. +1 in 2 VGPRs.
// B 4x16 f32 layout mirrors it with B[k][n] = V[n][k] -> same per-lane addressing.
__global__ void lgcn_gamma_wmma_kernel(float* __restrict__ out, const float* __restrict__ acc,
                                       const int* __restrict__ users, const int* __restrict__ items,
                                       int NU, int B) {
  __shared__ float tile[16 * 17];
  const int t = blockIdx.x;
  const int l = threadIdx.x;            // 0..31, always fully active before WMMA
  int mrow = t * 16 + (l & 15);
  if (mrow > B - 1) mrow = B - 1;       // clamp (keep EXEC all-ones for WMMA)
  const int half = l >> 4;
  const float* up = acc + (size_t)users[mrow] * DIMF;
  const float* vp = acc + ((size_t)NU + (size_t)items[mrow]) * DIMF;

  v8f c = {0.f, 0.f, 0.f, 0.f, 0.f, 0.f, 0.f, 0.f};
#pragma unroll
  for (int k0 = 0; k0 < DIMF; k0 += 4) {
    const int col = k0 + 2 * half;
    v2f a; a.x = up[col]; a.y = up[col + 1];
    v2f b; b.x = vp[col]; b.y = vp[col + 1];
    c = __builtin_amdgcn_wmma_f32_16x16x4_f32(false, a, false, b, (short)0, c, false, false);
  }

  // D layout: VGPR r -> M = r + 8*(l>>4), N = l&15. Stage tile, read diagonal.
#pragma unroll
  for (int r = 0; r < 8; ++r)
    tile[(r + 8 * half) * 17 + (l & 15)] = c[r];
  __syncthreads();

  if (l < 16) {
    const int p = t * 16 + l;
    if (p < B) out[p] = tile[l * 17 + l] * (1.0f / 16.0f);  // mean/4 on each side
  }
}

extern "C" void kernel_launch(void* const* d_in, const int* in_sizes, int n_in,
                              void* d_out, int out_size, void* d_ws, size_t ws_size,
                              hipStream_t stream) {
  const int*   users     = (const int*)d_in[0];
  const int*   items     = (const int*)d_in[1];
  const int*   rows      = (const int*)d_in[2];
  const int*   cols      = (const int*)d_in[3];
  const float* vals      = (const float*)d_in[4];
  const float* user_emb  = (const float*)d_in[5];
  const float* item_emb  = (const float*)d_in[6];
  const float* user_emb0 = (const float*)d_in[7];
  const float* item_emb0 = (const float*)d_in[8];
  float* out = (float*)d_out;

  const int B   = in_sizes[0];
  const int nnz = in_sizes[2];
  const int NU  = in_sizes[5] / DIMF;
  const int NI  = in_sizes[6] / DIMF;
  const int N   = NU + NI;

  const size_t nd = (size_t)N * DIMF;
  float* bufA = (float*)d_ws;          // ping
  float* bufB = bufA + nd;             // pong
  float* acc  = bufB + nd;             // weighted sum accumulator

  const int blk = 256;
  const long total4 = (long)(nd / 4);
  const unsigned gElem = (unsigned)((total4 + blk - 1) / blk);
  const unsigned gSpmm = (unsigned)(((long)nnz * 16 + blk - 1) / blk);

  lgcn_init_kernel<<<gElem, blk, 0, stream>>>(bufA, acc, user_emb, item_emb, NU, total4);

  float* cur = bufA;
  float* nxt = bufB;
  const float w[3] = {3.f, 2.f, 1.f};
  for (int L = 0; L < 3; ++L) {
    lgcn_layerinit_kernel<<<gElem, blk, 0, stream>>>(nxt, user_emb0, item_emb0, NU, total4);
    lgcn_spmm_kernel<<<gSpmm, blk, 0, stream>>>(nxt, cur, rows, cols, vals, nnz);
    lgcn_accum_kernel<<<gElem, blk, 0, stream>>>(acc, nxt, w[L], total4);
    float* tmp = cur; cur = nxt; nxt = tmp;
  }

  const unsigned ntile = (unsigned)((B + 15) / 16);
  lgcn_gamma_wmma_kernel<<<ntile, 32, 0, stream>>>(out, acc, users, items, NU, B);
}